// Atten_Model_20083267076674
// MI455X (gfx1250) — compile-verified
//
#include <hip/hip_runtime.h>

#define NN   8192
#define INF  512
#define HH   64
#define DEG  32
#define EE   (NN * DEG)

typedef __attribute__((ext_vector_type(2))) float v2f;
typedef __attribute__((ext_vector_type(8))) float v8f;
typedef __attribute__((ext_vector_type(4))) float f4;

// ---------------------------------------------------------------------------
// Kernel 1: Wx = x @ W  via fp32 WMMA (V_WMMA_F32_16X16X4_F32), one wave per
// 16x16 output tile, K looped in steps of 4.
// A (16x4 f32): lanes 0-15 -> M=lane, v0=K0,v1=K1 ; lanes 16-31 -> v0=K2,v1=K3
// B (4x16 f32): lanes 0-15 -> N=lane, v0=K0,v1=K1 ; lanes 16-31 -> v0=K2,v1=K3
// D (16x16 f32): vgpr r -> M = r + 8*(lane>>4), N = lane&15
// ---------------------------------------------------------------------------
__global__ void gat_gemm_wmma(const float* __restrict__ x,
                              const float* __restrict__ W,
                              float* __restrict__ Wx) {
  const int wid   = threadIdx.x >> 5;
  const int gid   = blockIdx.x * (blockDim.x >> 5) + wid;
  const int tileM = gid >> 2;          // 512 M-tiles
  const int tileN = gid & 3;           // 4 N-tiles
  const int lane  = threadIdx.x & 31;
  const int lo    = lane & 15;
  const int hi    = lane >> 4;         // 0 or 1

  const float* xrow = x + (size_t)(tileM * 16 + lo) * INF;
  const int    col  = tileN * 16 + lo;

  v8f c = {0.f, 0.f, 0.f, 0.f, 0.f, 0.f, 0.f, 0.f};

  for (int k = 0; k < INF; k += 4) {
    v2f a = *(const v2f*)(xrow + k + 2 * hi);          // contiguous pair, 8B aligned
    v2f b;
    b.x = W[(k + 2 * hi)     * HH + col];
    b.y = W[(k + 2 * hi + 1) * HH + col];
    c = __builtin_amdgcn_wmma_f32_16x16x4_f32(false, a, false, b,
                                              (short)0, c, false, false);
  }

#pragma unroll
  for (int r = 0; r < 8; ++r) {
    const int rowOut = tileM * 16 + r + 8 * hi;
    Wx[rowOut * HH + col] = c[r];
  }
}

// ---------------------------------------------------------------------------
// Kernel 2: s[i] = Wx[i,:] . a[0:64] ; t[i] = Wx[i,:] . a[64:128]
// One wave per row, 2 elements per lane, shfl_xor reduction (wave32).
// ---------------------------------------------------------------------------
__global__ void gat_proj(const float* __restrict__ Wx,
                         const float* __restrict__ a,
                         float* __restrict__ s,
                         float* __restrict__ t) {
  const int wid  = threadIdx.x >> 5;
  const int lane = threadIdx.x & 31;
  const int row  = blockIdx.x * (blockDim.x >> 5) + wid;

  const float* wr = Wx + (size_t)row * HH;
  const float w0 = wr[lane];
  const float w1 = wr[lane + 32];

  float ps = w0 * a[lane]      + w1 * a[lane + 32];
  float pt = w0 * a[HH + lane] + w1 * a[HH + lane + 32];

#pragma unroll
  for (int off = 16; off > 0; off >>= 1) {
    ps += __shfl_xor(ps, off, 32);
    pt += __shfl_xor(pt, off, 32);
  }
  if (lane == 0) { s[row] = ps; t[row] = pt; }
}

// ---------------------------------------------------------------------------
// Kernel 3a: per-row attention coefficients, normalized. One wave per row:
// lane j handles edge (row -> dst[row*32+j]). Row sum via shfl_xor.
// dflag[row] = 1 if row sum was zero (reference's diagonal fixup).
// ---------------------------------------------------------------------------
__global__ void gat_coef(const float* __restrict__ s,
                         const float* __restrict__ t,
                         const int* __restrict__ dstIdx,
                         float* __restrict__ coefN,
                         float* __restrict__ dflag) {
  const int wid  = threadIdx.x >> 5;
  const int lane = threadIdx.x & 31;
  const int row  = blockIdx.x * (blockDim.x >> 5) + wid;

  const int dst = dstIdx[row * DEG + lane];
  const float e  = s[row] + t[dst];
  const float lr = (e > 0.f) ? e : 0.1f * e;     // leaky_relu, slope 0.1
  const float coef = expf(lr);

  float sum = coef;
#pragma unroll
  for (int off = 16; off > 0; off >>= 1) sum += __shfl_xor(sum, off, 32);

  const float val = (sum == 0.f) ? 0.f : coef / sum;
  coefN[row * DEG + lane] = val;
  if (lane == 0) dflag[row] = (sum == 0.f) ? 1.f : 0.f;
}

// ---------------------------------------------------------------------------
// Kernel 3b: stream out the dense row-normalized adjacency (256 MB).
// One block per row; thread t owns 32 consecutive columns (8 x b128 stores).
// Fast path (almost all threads): nontemporal zero stores.
// Slow path: reconstruct value per column: d=(c-row-1) mod N; d<32 -> coef,
// c==row -> dflag, else 0. Single write per output element => no ordering
// hazards, minimal bandwidth, NT hint since 256MB >> 192MB L2.
// ---------------------------------------------------------------------------
__global__ void gat_write(const float* __restrict__ coefN,
                          const float* __restrict__ dflag,
                          float* __restrict__ out) {
  const int row = blockIdx.x;
  const int tid = threadIdx.x;
  const int c0  = tid * 32;
  float* rp = out + (size_t)row * NN + c0;

  // d-values covered by this chunk: d0 .. d0+31 (mod N)
  const int d0 = (c0 - row - 1) & (NN - 1);
  const bool special = (d0 < DEG) || (d0 >= NN - 31);
  const bool hasdiag = (row >= c0) && (row < c0 + 32);

  if (!special && !hasdiag) {
    const f4 z = {0.f, 0.f, 0.f, 0.f};
#pragma unroll
    for (int q = 0; q < 8; ++q)
      __builtin_nontemporal_store(z, (f4*)(rp + q * 4));
  } else {
    const float* cr = coefN + (size_t)row * DEG;
    const float df = dflag[row];
#pragma unroll
    for (int q = 0; q < 8; ++q) {
      f4 v;
#pragma unroll
      for (int j = 0; j < 4; ++j) {
        const int c = c0 + q * 4 + j;
        const int d = (c - row - 1) & (NN - 1);
        float xv = 0.f;
        if (d < DEG)       xv = cr[d];
        else if (c == row) xv = df;
        v[j] = xv;
      }
      __builtin_nontemporal_store(v, (f4*)(rp + q * 4));
    }
  }
}

// ---------------------------------------------------------------------------
extern "C" void kernel_launch(void* const* d_in, const int* in_sizes, int n_in,
                              void* d_out, int out_size, void* d_ws, size_t ws_size,
                              hipStream_t stream) {
  const float* x    = (const float*)d_in[0];   // [N, IN]
  const float* W    = (const float*)d_in[1];   // [IN, H]
  const float* a    = (const float*)d_in[2];   // [2H, 1]
  const int*   edge = (const int*)d_in[3];     // [2, E]
  float* out = (float*)d_out;                  // [N, N]

  float* Wx    = (float*)d_ws;                 // N*H      = 2 MB
  float* s     = Wx + (size_t)NN * HH;         // N
  float* t     = s + NN;                       // N
  float* coefN = t + NN;                       // N*DEG    = 1 MB
  float* dflag = coefN + (size_t)NN * DEG;     // N

  const int* dstIdx = edge + EE;               // edge_index[1]

  // 1) Wx = x @ W  (2048 waves, one 16x16 tile each)
  gat_gemm_wmma<<<256, 256, 0, stream>>>(x, W, Wx);
  // 2) per-node projections s, t
  gat_proj<<<NN / 8, 256, 0, stream>>>(Wx, a, s, t);
  // 3a) normalized coefficients per row
  gat_coef<<<NN / 8, 256, 0, stream>>>(s, t, dstIdx, coefN, dflag);
  // 3b) stream the dense normalized adjacency
  gat_write<<<NN, 256, 0, stream>>>(coefN, dflag, out);
}